// Linear_4bit_73667279061291
// MI455X (gfx1250) — compile-verified
//
#include <hip/hip_runtime.h>

// y = W @ x + b : 8192x8192 fp32 matvec. Pure HBM-bandwidth problem
// (0.5 flop/byte). Strategy: K-split x8 for wave-level parallelism,
// async global->LDS double-buffered staging of W, V_WMMA_F32_16X16X4_F32
// for the multiply (16 output rows per wave, x in column 0 of B),
// deterministic two-pass reduction of K-partials through d_ws.

#define IN_F  8192
#define OUT_F 8192

constexpr int KSPLIT  = 8;                 // K-dimension split across blockIdx.y
constexpr int KRANGE  = IN_F / KSPLIT;     // 1024 columns per WG
constexpr int KT      = 64;                // K-tile staged in LDS per stage
constexpr int STAGES  = KRANGE / KT;       // 16
constexpr int ROWS_WG = 128;               // 8 waves * 16 rows
constexpr int WSTRIDE = KT + 4;            // 68 dwords: even (b64-aligned), 68%64=4 -> conflict-free

typedef float v2f __attribute__((ext_vector_type(2)));
typedef float v4f __attribute__((ext_vector_type(4)));
typedef float v8f __attribute__((ext_vector_type(8)));
typedef int   v4i __attribute__((ext_vector_type(4)));

typedef __attribute__((address_space(1))) v4i* gv4i_p;  // global
typedef __attribute__((address_space(3))) v4i* lv4i_p;  // LDS

// Device-pass sanity check only (builtins are invisible to the host pass).
#if defined(__HIP_DEVICE_COMPILE__) && !__has_builtin(__builtin_amdgcn_wmma_f32_16x16x4_f32)
#error "wmma_f32_16x16x4_f32 builtin not available on this toolchain"
#endif

// One 16-byte async DMA lane: global -> LDS, tracked by ASYNCcnt.
__device__ __forceinline__ void async_copy16(const float* g, float* l) {
#if __has_builtin(__builtin_amdgcn_global_load_async_to_lds_b128)
  __builtin_amdgcn_global_load_async_to_lds_b128((gv4i_p)g, (lv4i_p)l, 0, 0);
#else
  *(v4f*)l = *(const v4f*)g;   // host-pass placeholder / sync fallback
#endif
}

__device__ __forceinline__ void wait_async(int n) {
#if __has_builtin(__builtin_amdgcn_s_wait_asynccnt)
  if (n == 0) __builtin_amdgcn_s_wait_asynccnt(0);
  else        __builtin_amdgcn_s_wait_asynccnt(8);
#endif
}

__device__ __forceinline__ v8f wmma_16x16x4_f32(v2f a, v2f b, v8f c) {
#if __has_builtin(__builtin_amdgcn_wmma_f32_16x16x4_f32)
  return __builtin_amdgcn_wmma_f32_16x16x4_f32(false, a, false, b,
                                               (short)0, c, false, false);
#else
  return c;  // host-pass placeholder, never executed
#endif
}

__global__ void __launch_bounds__(256)
matvec_wmma(const float* __restrict__ W, const float* __restrict__ x,
            float* __restrict__ dst, int atomic_mode) {
  // xls FIRST: keeps x addresses within the 16-bit DS immediate-offset field,
  // so the inner-loop x loads need no per-iteration address arithmetic.
  __shared__ __align__(16) float xls[KRANGE];                // 4 KB @ LDS offset 0
  __shared__ __align__(16) float wls[2][ROWS_WG * WSTRIDE];  // 2 x 34 KB

  const int tid     = threadIdx.x;
  const int lane    = tid & 31;
  const int wave    = tid >> 5;
  const int rowBase = blockIdx.x * ROWS_WG;
  const int kBase   = blockIdx.y * KRANGE;

  const int rr = tid >> 4;          // staging: row-in-round   (0..15)
  const int cc = (tid & 15) * 4;    // staging: col (floats)   (0,4,..,60)

  // ---- prologue: x slice (KRANGE floats) + stage-0 W tile ----
  async_copy16(x + kBase + tid * 4, &xls[tid * 4]);
  {
    const float* gw = W + (size_t)rowBase * IN_F + kBase;
#pragma unroll
    for (int rd = 0; rd < ROWS_WG / 16; ++rd) {       // 8 issues/thread
      const int r = rd * 16 + rr;
      async_copy16(gw + (size_t)r * IN_F + cc, &wls[0][r * WSTRIDE + cc]);
    }
  }

  // WMMA A-layout (16x4 f32): lane L<16 holds row M=L, K=0..1;
  // lane L>=16 holds row M=L-16, K=2..3  -> one ds_load_b64 per step.
  const int mrow = (wave << 4) + (lane & 15);
  const int kh   = (lane >> 4) << 1;                  // 0 or 2
  const bool col0 = ((lane & 15) == 0);               // B column N=0 lanes (0 and 16)

  v8f acc0 = {};
  v8f acc1 = {};

  for (int s = 0; s < STAGES; ++s) {
    if (s + 1 < STAGES) {                             // issue next stage (double buffer)
      float*       buf = wls[(s + 1) & 1];
      const float* gw  = W + (size_t)rowBase * IN_F + kBase + (s + 1) * KT;
#pragma unroll
      for (int rd = 0; rd < ROWS_WG / 16; ++rd) {
        const int r = rd * 16 + rr;
        async_copy16(gw + (size_t)r * IN_F + cc, buf + r * WSTRIDE + cc);
      }
      wait_async(8);                                  // drain everything except the 8 just issued
    } else {
      wait_async(0);
    }
    __syncthreads();                                  // stage s visible to all waves

    const float* wrow = &wls[s & 1][mrow * WSTRIDE + kh];
    const float* xrow = &xls[s * KT + kh];
#pragma unroll
    for (int k = 0; k < KT; k += 8) {
      v2f a0  = *(const v2f*)(wrow + k);              // ds_load_2addr_b64, conflict-free
      v2f xv0 = *(const v2f*)(xrow + k);              // LDS broadcast, imm offset
      v2f b0  = { col0 ? xv0.x : 0.0f, col0 ? xv0.y : 0.0f };
      acc0 = wmma_16x16x4_f32(a0, b0, acc0);
      v2f a1  = *(const v2f*)(wrow + k + 4);
      v2f xv1 = *(const v2f*)(xrow + k + 4);
      v2f b1  = { col0 ? xv1.x : 0.0f, col0 ? xv1.y : 0.0f };
      acc1 = wmma_16x16x4_f32(a1, b1, acc1);
    }
    __syncthreads();                                  // protect buffer before next-stage DMA lands
  }

  // D column N=0: lane 0 holds M=0..7 in v[0..7], lane 16 holds M=8..15.
  v8f acc = acc0 + acc1;
  if (col0) {
    const int r0 = rowBase + (wave << 4) + ((lane >> 4) << 3);
    if (atomic_mode) {
#pragma unroll
      for (int r = 0; r < 8; ++r) atomicAdd(&dst[r0 + r], acc[r]);
    } else {
      float* p = dst + (size_t)blockIdx.y * OUT_F + r0;   // 32B-aligned
      *(v4f*)(p + 0) = v4f{acc[0], acc[1], acc[2], acc[3]};
      *(v4f*)(p + 4) = v4f{acc[4], acc[5], acc[6], acc[7]};
    }
  }
}

__global__ void __launch_bounds__(256)
reduce_bias(const float* __restrict__ ws, const float* __restrict__ bias,
            float* __restrict__ out) {
  const int o = blockIdx.x * 256 + threadIdx.x;
  float s = bias[o];
#pragma unroll
  for (int j = 0; j < KSPLIT; ++j) s += ws[(size_t)j * OUT_F + o];
  out[o] = s;
}

__global__ void __launch_bounds__(256)
copy_bias(const float* __restrict__ bias, float* __restrict__ out) {
  const int o = blockIdx.x * 256 + threadIdx.x;
  out[o] = bias[o];
}

extern "C" void kernel_launch(void* const* d_in, const int* in_sizes, int n_in,
                              void* d_out, int out_size, void* d_ws, size_t ws_size,
                              hipStream_t stream) {
  const float* x    = (const float*)d_in[0];   // input  (8192)
  const float* W    = (const float*)d_in[1];   // weight (8192x8192)
  const float* bias = (const float*)d_in[2];   // bias   (8192)
  float* out = (float*)d_out;

  const dim3 grid(OUT_F / ROWS_WG, KSPLIT);    // 64 x 8 = 512 WGs, 4096 waves
  const size_t ws_need = (size_t)KSPLIT * OUT_F * sizeof(float);  // 256 KB

  if (ws_size >= ws_need) {
    // Deterministic two-pass: partials -> d_ws, then reduce + bias.
    matvec_wmma<<<grid, 256, 0, stream>>>(W, x, (float*)d_ws, 0);
    reduce_bias<<<OUT_F / 256, 256, 0, stream>>>((const float*)d_ws, bias, out);
  } else {
    // Fallback: seed out with bias, accumulate partials atomically.
    copy_bias<<<OUT_F / 256, 256, 0, stream>>>(bias, out);
    matvec_wmma<<<grid, 256, 0, stream>>>(W, x, out, 1);
  }
}